// MagPred_49821620634213
// MI455X (gfx1250) — compile-verified
//
#include <hip/hip_runtime.h>
#include <hip/hip_bf16.h>

typedef __attribute__((ext_vector_type(2))) float v2f;
typedef __attribute__((ext_vector_type(8))) float v8f;

#define N_SRC  2048
#define N_GRID 50000
#define N_LOCS 500
#define KTOP   5
#define WAVES_PER_BLOCK 8
#define N_TILES (N_GRID / 16)                                   // 3125, exact
#define TILES_PER_WAVE ((N_TILES + WAVES_PER_BLOCK - 1) / WAVES_PER_BLOCK)  // 391

// ---------------------------------------------------------------------------
// Kernel 1: fused distance-eval (WMMA f32 16x16x4) + streaming per-source top-5
//   rank key: |g|^2 - 2*s.g   (== d2 minus the rank-invariant |s|^2, unscaled)
//   A (16x4, M=grid tile):  lane<16:  {gx, gy} = K0,K1 for row M=lane
//                           lane>=16: {gz, gg} = K2,K3 for row M=lane-16
//   B (4x16, N=sources):    lane<16:  {-2sx, -2sy} = K0,K1 for col N=lane
//                           lane>=16: {-2sz, 1.0 } = K2,K3 for col N=lane-16
//   D (16x16): lane<16 holds col N=lane rows 0..7; lane>=16 rows 8..15
// Tiles (3125 = 50000/16 exactly) are distributed whole, so no OOB rows, no
// guards, no duplicates, and EXEC is all-ones at every WMMA. The tile load is
// software-pipelined one iteration ahead so s_wait_loadcnt lands after the
// WMMA + filter work instead of immediately after issue.
// ---------------------------------------------------------------------------
__global__ __launch_bounds__(32 * WAVES_PER_BLOCK)
void topk_kernel(const float* __restrict__ src,
                 const float* __restrict__ grid,
                 int* __restrict__ inds_out) {
    __shared__ float s_val[WAVES_PER_BLOCK * 32 * KTOP];
    __shared__ int   s_idx[WAVES_PER_BLOCK * 32 * KTOP];

    const int tid  = threadIdx.x;
    const int wave = tid >> 5;
    const int lane = tid & 31;
    const int half = lane >> 4;     // 0: lanes 0-15, 1: lanes 16-31
    const int l16  = lane & 15;
    const int sbase = blockIdx.x * 16;

    // B operand: loop-invariant, one source per lane column
    const float* sp = src + (size_t)(sbase + l16) * 3;
    const float sx = sp[0], sy = sp[1], sz = sp[2];
    v2f bmat;
    bmat.x = half ? (-2.0f * sz) : (-2.0f * sx);
    bmat.y = half ? 1.0f         : (-2.0f * sy);

    float topv[KTOP];
    int   topi[KTOP];
#pragma unroll
    for (int k = 0; k < KTOP; ++k) { topv[k] = 3.0e38f; topi[k] = 0; }

    const int tbeg = wave * TILES_PER_WAVE;
    const int tend = (tbeg + TILES_PER_WAVE < N_TILES) ? (tbeg + TILES_PER_WAVE)
                                                       : N_TILES;

    // software pipeline: prologue load of the first tile
    const float* gp0 = grid + ((size_t)(tbeg * 16 + l16)) * 3;
    float gx = gp0[0], gy = gp0[1], gz = gp0[2];

    for (int t = tbeg; t < tend; ++t) {
        // issue next tile's load now (clamped: last iteration reloads itself,
        // uniform scalar select, never out of bounds)
        const int tn = (t + 1 < tend) ? (t + 1) : t;
        const float* gn = grid + ((size_t)(tn * 16 + l16)) * 3;
        const float nx = gn[0], ny = gn[1], nz = gn[2];
        // stream further ahead toward the WGP (speculative; OOB dropped)
        __builtin_prefetch(gn + 96, 0, 3);

        // process current tile while the next load is in flight
        const float gg = fmaf(gx, gx, fmaf(gy, gy, gz * gz));
        v2f amat;
        amat.x = half ? gz : gx;
        amat.y = half ? gg : gy;
        v8f c = {};
        v8f d = __builtin_amdgcn_wmma_f32_16x16x4_f32(
            /*neg_a=*/false, amat, /*neg_b=*/false, bmat,
            /*c_mod=*/(short)0, c, /*reuse_a=*/false, /*reuse_b=*/false);

        const int mbase = t * 16 + half * 8;
#pragma unroll
        for (int r = 0; r < 8; ++r) {
            const float val = d[r];
            if (val < topv[KTOP - 1]) {
                topv[KTOP - 1] = val;
                topi[KTOP - 1] = mbase + r;
#pragma unroll
                for (int j = KTOP - 1; j > 0; --j) {
                    if (topv[j] < topv[j - 1]) {
                        float tv = topv[j]; topv[j] = topv[j - 1]; topv[j - 1] = tv;
                        int   ti = topi[j]; topi[j] = topi[j - 1]; topi[j - 1] = ti;
                    }
                }
            }
        }

        // rotate pipeline registers (first use of the in-flight load)
        gx = nx; gy = ny; gz = nz;
    }

    // stage per-lane top-5s
    const int base = (wave * 32 + lane) * KTOP;
#pragma unroll
    for (int k = 0; k < KTOP; ++k) { s_val[base + k] = topv[k]; s_idx[base + k] = topi[k]; }
    __syncthreads();

    // final merge: one thread per source merges 8 waves x 2 half-lanes x 5 = 80 cands
    if (tid < 16) {
        float bv[KTOP];
        int   bi[KTOP];
#pragma unroll
        for (int k = 0; k < KTOP; ++k) { bv[k] = 3.0e38f; bi[k] = 0; }
        for (int w = 0; w < WAVES_PER_BLOCK; ++w) {
            for (int h = 0; h < 2; ++h) {
                const int lb = (w * 32 + h * 16 + tid) * KTOP;
#pragma unroll
                for (int k = 0; k < KTOP; ++k) {
                    const float val = s_val[lb + k];
                    const int   m   = s_idx[lb + k];
                    if (val < bv[KTOP - 1]) {
                        bv[KTOP - 1] = val;
                        bi[KTOP - 1] = m;
#pragma unroll
                        for (int j = KTOP - 1; j > 0; --j) {
                            if (bv[j] < bv[j - 1]) {
                                float tv = bv[j]; bv[j] = bv[j - 1]; bv[j - 1] = tv;
                                int   ti = bi[j]; bi[j] = bi[j - 1]; bi[j - 1] = ti;
                            }
                        }
                    }
                }
            }
        }
        const int n = sbase + tid;
#pragma unroll
        for (int k = 0; k < KTOP; ++k) inds_out[n * KTOP + k] = bi[k];
    }
}

// ---------------------------------------------------------------------------
// Kernel 2: bias gather-mean + log10 epilogue, one block per source
// ---------------------------------------------------------------------------
__global__ __launch_bounds__(256)
void mag_kernel(const float* __restrict__ src,
                const int*   __restrict__ ind,
                const float* __restrict__ log_amp,
                const int*   __restrict__ phase_p,
                const float* __restrict__ locs,
                const float* __restrict__ mag_coef,
                const float* __restrict__ eps_coef,
                const float* __restrict__ dep_coef,
                const float* __restrict__ bias,
                const int*   __restrict__ inds,
                float* __restrict__ out) {
    const int i  = blockIdx.x;
    const int ph = phase_p[0];
    const float ec    = eps_coef[ph];
    const float dc    = dep_coef[ph];
    const float denom = fmaxf(mag_coef[ph], 1e-12f);

    const float sx = src[(size_t)i * 3 + 0];
    const float sy = src[(size_t)i * 3 + 1];
    const float sz = src[(size_t)i * 3 + 2];

    int g[KTOP];
#pragma unroll
    for (int k = 0; k < KTOP; ++k) g[k] = inds[i * KTOP + k];

    for (int l = threadIdx.x; l < N_LOCS; l += blockDim.x) {
        const int li = ind[l];
        const float lx = locs[(size_t)li * 3 + 0];
        const float ly = locs[(size_t)li * 3 + 1];
        const float lz = locs[(size_t)li * 3 + 2];

        const float dx = sx - lx, dy = sy - ly;
        const float d0  = sqrtf(fmaf(dx, dx, dy * dy));
        const float pz  = log10f(d0 + 1.0f);                 // FUDGE = 1.0
        const float pd  = log10f(fabsf(sz - lz) + 1.0f);

        float b = 0.0f;
#pragma unroll
        for (int k = 0; k < KTOP; ++k)
            b += bias[((size_t)g[k] * N_LOCS + (size_t)li) * 2 + ph];
        b *= 0.2f;

        const float la = log_amp[(size_t)i * N_LOCS + l];
        out[(size_t)i * N_LOCS + l] = (la - ec * pz - dc * pd - b) / denom;
    }
}

extern "C" void kernel_launch(void* const* d_in, const int* in_sizes, int n_in,
                              void* d_out, int out_size, void* d_ws, size_t ws_size,
                              hipStream_t stream) {
    const float* src      = (const float*)d_in[0];   // (2048,3)
    const int*   ind      = (const int*)  d_in[1];   // (500,)
    const float* log_amp  = (const float*)d_in[2];   // (2048,500)
    const int*   phase    = (const int*)  d_in[3];   // scalar
    const float* locs     = (const float*)d_in[4];   // (500,3)
    const float* grid     = (const float*)d_in[5];   // (50000,3)
    const float* mag_coef = (const float*)d_in[6];   // (2,)
    const float* eps_coef = (const float*)d_in[7];   // (2,)
    const float* dep_coef = (const float*)d_in[8];   // (2,)
    const float* bias     = (const float*)d_in[9];   // (50000,500,2)

    int* inds = (int*)d_ws;                          // 2048*5 ints = 40 KB scratch

    topk_kernel<<<N_SRC / 16, 32 * WAVES_PER_BLOCK, 0, stream>>>(src, grid, inds);
    mag_kernel<<<N_SRC, 256, 0, stream>>>(src, ind, log_amp, phase, locs,
                                          mag_coef, eps_coef, dep_coef, bias,
                                          inds, (float*)d_out);
}